// Encoder_66314295050609
// MI455X (gfx1250) — compile-verified
//
#include <hip/hip_runtime.h>

// GraphSAGE 2-layer forward for MI455X (gfx1250, wave32).
//   layer(z): summed[dst] += relu(z[src]); mean = summed/max(deg,1)
//             out = mean @ Wl + bl + z @ Wr
// Scatter phase: L2-resident f32 atomics (features + accumulator fit in 192MB L2).
// Combine phase: fp32 WMMA (V_WMMA_F32_16X16X4_F32), fused dual-GEMM per 16-node tile.

typedef float v2f __attribute__((ext_vector_type(2)));
typedef float v8f __attribute__((ext_vector_type(8)));

#define D 128     // feature dim (D_IN == D_H == 128)
#define NTILE 16  // node rows per wave tile

// ---------------------------------------------------------------- zero fill
__global__ void zero_kernel(float* __restrict__ p, long n) {
  long i = (long)blockIdx.x * blockDim.x + threadIdx.x;
  long stride = (long)gridDim.x * blockDim.x;
  for (; i < n; i += stride) p[i] = 0.0f;
}

// ------------------------------------------------- edge scatter (1 wave/edge)
// Each wave: 32 lanes x float4 = 128 channels. relu then global f32 atomics.
__global__ void scatter_kernel(const float* __restrict__ z,
                               const int* __restrict__ src,
                               const int* __restrict__ dst,
                               float* __restrict__ summed,
                               float* __restrict__ cnt,  // nullptr => skip degree
                               int E) {
  unsigned tid = blockIdx.x * blockDim.x + threadIdx.x;
  int wid = (int)(tid >> 5);
  int lane = threadIdx.x & 31;
  if (wid >= E) return;
  int s = __builtin_amdgcn_readfirstlane(src[wid]);
  int d = __builtin_amdgcn_readfirstlane(dst[wid]);

  const float4* zp = (const float4*)(z + (size_t)s * D);
  float4 v = zp[lane];

  float* op = summed + (size_t)d * D + lane * 4;
  unsafeAtomicAdd(op + 0, fmaxf(v.x, 0.0f));
  unsafeAtomicAdd(op + 1, fmaxf(v.y, 0.0f));
  unsafeAtomicAdd(op + 2, fmaxf(v.z, 0.0f));
  unsafeAtomicAdd(op + 3, fmaxf(v.w, 0.0f));
  if (cnt != nullptr && lane == 0) unsafeAtomicAdd(cnt + d, 1.0f);
}

// -------------------------------------------------- deg -> 1/max(deg,1)
__global__ void inv_kernel(float* __restrict__ cnt, int n) {
  int i = blockIdx.x * blockDim.x + threadIdx.x;
  if (i < n) cnt[i] = 1.0f / fmaxf(cnt[i], 1.0f);
}

// ---------------------------------------------------------------- combine
// out[tile rows] = (summed*inv) @ Wl + bl + z @ Wr   (+ optional relu)
// One wave per 16-node tile; 8 f32 WMMA accumulators cover 128 out channels.
__global__ void __launch_bounds__(256)
combine_kernel(const float* __restrict__ summed, const float* __restrict__ inv,
               const float* __restrict__ z, const float* __restrict__ Wl,
               const float* __restrict__ bl, const float* __restrict__ Wr,
               float* __restrict__ out, int ntiles, int do_relu) {
  int tile = blockIdx.x * (blockDim.x >> 5) + (threadIdx.x >> 5);
  if (tile >= ntiles) return;  // wave-uniform: EXEC stays all-ones for WMMA
  int lane = threadIdx.x & 31;
  int lrow  = lane & 15;        // A row in tile / B,C column
  int khalf = (lane >> 4) * 2;  // lanes 16-31 handle K+2,K+3

  v8f acc[8] = {};

  int row = tile * NTILE + lrow;
  float invr = inv[row];
  const float* arow = summed + (size_t)row * D;
  const float* zrow = z + (size_t)row * D;

  // GEMM 1: mean @ Wl  (mean scaled on the fly)
  for (int k0 = 0; k0 < D; k0 += 4) {
    v2f a;
    a.x = arow[k0 + khalf + 0] * invr;
    a.y = arow[k0 + khalf + 1] * invr;
    const float* wp = Wl + (size_t)(k0 + khalf) * D + lrow;
#pragma unroll
    for (int n = 0; n < 8; ++n) {
      v2f b;
      b.x = wp[n * 16];
      b.y = wp[n * 16 + D];
      acc[n] = __builtin_amdgcn_wmma_f32_16x16x4_f32(
          false, a, false, b, (short)0, acc[n], false, false);
    }
  }
  // GEMM 2: z @ Wr  (same accumulators)
  for (int k0 = 0; k0 < D; k0 += 4) {
    v2f a;
    a.x = zrow[k0 + khalf + 0];
    a.y = zrow[k0 + khalf + 1];
    const float* wp = Wr + (size_t)(k0 + khalf) * D + lrow;
#pragma unroll
    for (int n = 0; n < 8; ++n) {
      v2f b;
      b.x = wp[n * 16];
      b.y = wp[n * 16 + D];
      acc[n] = __builtin_amdgcn_wmma_f32_16x16x4_f32(
          false, a, false, b, (short)0, acc[n], false, false);
    }
  }

  // epilogue: bias (+relu) and store. D layout: VGPR v -> rows v / v+8, col lane&15.
  int mbase = tile * NTILE + ((lane >> 4) << 3);
#pragma unroll
  for (int n = 0; n < 8; ++n) {
    float bias = bl[n * 16 + lrow];
#pragma unroll
    for (int v = 0; v < 8; ++v) {
      float val = acc[n][v] + bias;
      if (do_relu) val = fmaxf(val, 0.0f);
      out[(size_t)(mbase + v) * D + n * 16 + lrow] = val;
    }
  }
}

extern "C" void kernel_launch(void* const* d_in, const int* in_sizes, int n_in,
                              void* d_out, int out_size, void* d_ws, size_t ws_size,
                              hipStream_t stream) {
  const float* x   = (const float*)d_in[0];
  const int*   ei  = (const int*)d_in[1];  // [2,E] int32
  const float* Wl1 = (const float*)d_in[2];
  const float* bl1 = (const float*)d_in[3];
  const float* Wr1 = (const float*)d_in[4];
  const float* Wl2 = (const float*)d_in[5];
  const float* bl2 = (const float*)d_in[6];
  const float* Wr2 = (const float*)d_in[7];
  float* out = (float*)d_out;

  const int N = in_sizes[0] / D;  // 100000
  const int E = in_sizes[1] / 2;  // 1600000
  const int ntiles = (N + NTILE - 1) / NTILE;  // 6250 (exact)

  const int* src = ei;
  const int* dst = ei + E;

  float* cnt = (float*)d_ws;     // N floats
  float* summed = cnt + N;       // N*D floats

  int scatter_blocks = (int)(((long)E * 32 + 255) / 256);
  int combine_blocks = (ntiles + 7) / 8;  // 8 waves (tiles) per block

  // ---- layer 1 ----
  zero_kernel<<<2048, 256, 0, stream>>>(cnt, (long)N * (D + 1));
  scatter_kernel<<<scatter_blocks, 256, 0, stream>>>(x, src, dst, summed, cnt, E);
  inv_kernel<<<(N + 255) / 256, 256, 0, stream>>>(cnt, N);
  // h = relu(combine) staged in d_out
  combine_kernel<<<combine_blocks, 256, 0, stream>>>(summed, cnt, x, Wl1, bl1, Wr1,
                                                     out, ntiles, 1);
  // ---- layer 2 ----
  zero_kernel<<<2048, 256, 0, stream>>>(summed, (long)N * D);
  scatter_kernel<<<scatter_blocks, 256, 0, stream>>>(out, src, dst, summed, nullptr, E);
  // in-place: each wave reads only its own rows of h before overwriting them
  combine_kernel<<<combine_blocks, 256, 0, stream>>>(summed, cnt, out, Wl2, bl2, Wr2,
                                                     out, ntiles, 0);
}